// CannyEdgeDetector_27238682591978
// MI455X (gfx1250) — compile-verified
//
#include <hip/hip_runtime.h>
#include <math.h>

#define IMG_H 1024
#define IMG_W 1024
#define IMG_B 8
#define HW (IMG_H * IMG_W)
#define LOW_THR 0.1f
#define HIGH_THR 0.2f
#define PI_F 3.14159265358979323846f

typedef __attribute__((ext_vector_type(4))) unsigned int u32x4;
typedef __attribute__((ext_vector_type(8))) int i32x8;
typedef __attribute__((ext_vector_type(4))) int i32x4;

// ---------------------------------------------------------------------------
// TDM: async DMA of a 2-D fp32 tile (lh rows x lw elems, row stride in elems)
// from global memory into LDS. Built per CDNA5 ISA ch.8 descriptor layout.
// Caller must be a single wave; follow with s_wait_tensorcnt + barrier.
// This toolchain (clang-23/therock-10.0) uses the 6-arg builtin form.
// ---------------------------------------------------------------------------
__device__ __forceinline__ void tdm_load_tile_f32(const float* gsrc, void* lds_dst,
                                                  int lw, int lh, int row_stride) {
  unsigned lds_base = (unsigned)(unsigned long long)lds_dst; // LDS byte offset (low 32)
  unsigned long long ga = (unsigned long long)gsrc;
  u32x4 g0;
  g0[0] = 1u;                                                 // count=1, user descriptor
  g0[1] = lds_base;                                           // lds_addr
  g0[2] = (unsigned)(ga & 0xFFFFFFFFull);                     // global_addr[31:0]
  g0[3] = (unsigned)((ga >> 32) & 0x01FFFFFFull) | (2u << 30);// addr[56:32] | type=2
  const unsigned td0 = (unsigned)lw;   // tensor_dim0 == tile_dim0 (pre-clamped)
  const unsigned td1 = (unsigned)lh;   // tensor_dim1 == tile_dim1
  const unsigned long long st0 = (unsigned long long)row_stride;
  i32x8 g1;
  g1[0] = (int)(2u << 16);                         // data_size = 4 bytes; no pad/iterate
  g1[1] = (int)(td0 << 16);                        // barrier_addr=0 | tensor_dim0 lo
  g1[2] = (int)((td0 >> 16) | (td1 << 16));        // tensor_dim0 hi | tensor_dim1 lo
  g1[3] = (int)((td1 >> 16) | (td0 << 16));        // tensor_dim1 hi | tile_dim0
  g1[4] = (int)td1;                                // tile_dim1 | tile_dim2=0
  g1[5] = (int)(unsigned)(st0 & 0xFFFFFFFFull);    // tensor_dim0_stride lo
  g1[6] = (int)(unsigned)((st0 >> 32) & 0xFFFFull);// stride hi | dim1_stride lo (0)
  g1[7] = 0;
  i32x4 gz4 = {0, 0, 0, 0};                        // groups 2/3: 2-D tensor, unused
  i32x8 gz8 = {0, 0, 0, 0, 0, 0, 0, 0};            // extra group (clang-23 form), zero
  __builtin_amdgcn_tensor_load_to_lds(g0, g1, gz4, gz4, gz8, 0);
}

// ---------------------------------------------------------------------------
// Pass 1: RGB -> grayscale
// ---------------------------------------------------------------------------
__global__ __launch_bounds__(256) void k_gray(const float* __restrict__ x,
                                              float* __restrict__ g) {
  const size_t i = (size_t)blockIdx.x * 256 + threadIdx.x;
  const size_t b = i / (size_t)HW;
  const size_t p = i - b * (size_t)HW;
  const float* px = x + b * 3 * (size_t)HW + p;
  g[i] = 0.299f * px[0] + 0.587f * px[(size_t)HW] + 0.114f * px[2 * (size_t)HW];
}

// ---------------------------------------------------------------------------
// Pass 2/3: 5x5 correlation (zero-padded), tile 64x16 + halo 2 staged via TDM
// ---------------------------------------------------------------------------
__global__ __launch_bounds__(256) void k_conv5(const float* __restrict__ in,
                                               const float* __restrict__ k5,
                                               float* __restrict__ out,
                                               int transpose) {
  __shared__ float s_in[20 * 68];
  __shared__ float s_k[25];
  const int tx0 = blockIdx.x * 64;
  const int ty0 = blockIdx.y * 16;
  const int b = blockIdx.z;
  const int x0c = max(tx0 - 2, 0), x1c = min(tx0 + 66, IMG_W);
  const int y0c = max(ty0 - 2, 0), y1c = min(ty0 + 18, IMG_H);
  const int lw = x1c - x0c, lh = y1c - y0c;
  if (threadIdx.x < 25) {
    const int a = threadIdx.x / 5, c = threadIdx.x % 5;
    s_k[threadIdx.x] = transpose ? k5[c * 5 + a] : k5[a * 5 + c];
  }
  if (threadIdx.x < 32) {  // wave 0 issues the DMA once per workgroup
    const float* src = in + (size_t)b * HW + (size_t)y0c * IMG_W + x0c;
    tdm_load_tile_f32(src, s_in, lw, lh, IMG_W);
    __builtin_amdgcn_s_wait_tensorcnt(0);
  }
  __syncthreads();
  for (int t = threadIdx.x; t < 64 * 16; t += 256) {
    const int lx = t & 63, ly = t >> 6;
    const int gx = tx0 + lx, gy = ty0 + ly;
    float acc = 0.f;
#pragma unroll
    for (int a = 0; a < 5; ++a) {
      const int yy = gy + a - 2;
#pragma unroll
      for (int c = 0; c < 5; ++c) {
        const int xx = gx + c - 2;
        float v = 0.f;
        if (yy >= 0 && yy < IMG_H && xx >= 0 && xx < IMG_W)
          v = s_in[(yy - y0c) * lw + (xx - x0c)];
        acc = fmaf(s_k[a * 5 + c], v, acc);
      }
    }
    out[(size_t)b * HW + (size_t)gy * IMG_W + gx] = acc;
  }
}

// ---------------------------------------------------------------------------
// Pass 4: Sobel + magnitude + direction-quantized NMS + double threshold
// ---------------------------------------------------------------------------
__global__ __launch_bounds__(256) void k_sobel_nms(const float* __restrict__ blur,
                                                   unsigned char* __restrict__ e,
                                                   unsigned char* __restrict__ lowm) {
  __shared__ float s_in[20 * 68];
  __shared__ float s_mag[18 * 66];
  const int tx0 = blockIdx.x * 64;
  const int ty0 = blockIdx.y * 16;
  const int b = blockIdx.z;
  const int x0c = max(tx0 - 2, 0), x1c = min(tx0 + 66, IMG_W);
  const int y0c = max(ty0 - 2, 0), y1c = min(ty0 + 18, IMG_H);
  const int lw = x1c - x0c, lh = y1c - y0c;
  if (threadIdx.x < 32) {
    const float* src = blur + (size_t)b * HW + (size_t)y0c * IMG_W + x0c;
    tdm_load_tile_f32(src, s_in, lw, lh, IMG_W);
    __builtin_amdgcn_s_wait_tensorcnt(0);
  }
  __syncthreads();
  auto ld = [&](int yy, int xx) -> float {
    return (yy >= 0 && yy < IMG_H && xx >= 0 && xx < IMG_W)
               ? s_in[(yy - y0c) * lw + (xx - x0c)]
               : 0.f;
  };
  // gradient magnitude over tile + halo 1
  for (int t = threadIdx.x; t < 18 * 66; t += 256) {
    const int gx = tx0 + (t % 66) - 1;
    const int gy = ty0 + (t / 66) - 1;
    const float a00 = ld(gy - 1, gx - 1), a01 = ld(gy - 1, gx), a02 = ld(gy - 1, gx + 1);
    const float a10 = ld(gy, gx - 1), a12 = ld(gy, gx + 1);
    const float a20 = ld(gy + 1, gx - 1), a21 = ld(gy + 1, gx), a22 = ld(gy + 1, gx + 1);
    const float gxv = (a02 - a00) + 2.f * (a12 - a10) + (a22 - a20);
    const float gyv = (a20 - a00) + 2.f * (a21 - a01) + (a22 - a02);
    s_mag[t] = sqrtf(gxv * gxv + gyv * gyv);
  }
  __syncthreads();
  for (int t = threadIdx.x; t < 64 * 16; t += 256) {
    const int lx = t & 63, ly = t >> 6;
    const int gx = tx0 + lx, gy = ty0 + ly;
    const float a00 = ld(gy - 1, gx - 1), a01 = ld(gy - 1, gx), a02 = ld(gy - 1, gx + 1);
    const float a10 = ld(gy, gx - 1), a12 = ld(gy, gx + 1);
    const float a20 = ld(gy + 1, gx - 1), a21 = ld(gy + 1, gx), a22 = ld(gy + 1, gx + 1);
    const float gxv = (a02 - a00) + 2.f * (a12 - a10) + (a22 - a20);
    const float gyv = (a20 - a00) + 2.f * (a21 - a01) + (a22 - a02);
    const float mag = s_mag[(ly + 1) * 66 + (lx + 1)];
    // cat = round_half_even(atan2/45deg) mod 4   (rintf == RNE, matches jnp.round)
    const int q = (int)rintf(atan2f(gyv, gxv) * (4.0f / PI_F));
    const int cat = ((q % 4) + 4) % 4;
    int d1y, d1x, d2y, d2x;
    if (cat == 0) { d1y = 0;  d1x = -1; d2y = 0; d2x = 1; }
    else if (cat == 1) { d1y = -1; d1x = 1;  d2y = 1; d2x = -1; }
    else if (cat == 2) { d1y = -1; d1x = 0;  d2y = 1; d2x = 0; }
    else { d1y = -1; d1x = -1; d2y = 1; d2x = 1; }
    const float n1 = s_mag[(ly + 1 + d1y) * 66 + (lx + 1 + d1x)];
    const float n2 = s_mag[(ly + 1 + d2y) * 66 + (lx + 1 + d2x)];
    const bool interior = (gx >= 1) && (gx < IMG_W - 1) && (gy >= 1) && (gy < IMG_H - 1);
    const float nms = (interior && mag >= fmaxf(n1, n2)) ? mag : 0.f;
    const size_t o = (size_t)b * HW + (size_t)gy * IMG_W + gx;
    e[o] = nms > HIGH_THR ? (unsigned char)1 : (unsigned char)0;
    lowm[o] = nms > LOW_THR ? (unsigned char)1 : (unsigned char)0;
  }
}

// ---------------------------------------------------------------------------
// Pass 5 (x16, ping-pong): hysteresis flood fill; block-local Gauss-Seidel to
// convergence inside a 64x64 LDS tile (monotone -> same fixed point as ref).
// ---------------------------------------------------------------------------
__global__ __launch_bounds__(256) void k_hyst(const unsigned char* __restrict__ ein,
                                              const unsigned char* __restrict__ lowm,
                                              unsigned char* __restrict__ eout) {
  __shared__ unsigned char s_e[66 * 66];
  __shared__ unsigned char s_low[64 * 64];
  __shared__ int s_changed;
  const int tx0 = blockIdx.x * 64;
  const int ty0 = blockIdx.y * 64;
  const int b = blockIdx.z;
  const unsigned char* eb = ein + (size_t)b * HW;
  for (int t = threadIdx.x; t < 66 * 66; t += 256) {
    const int gx = tx0 + (t % 66) - 1;
    const int gy = ty0 + (t / 66) - 1;
    unsigned char v = 0;
    if (gx >= 0 && gx < IMG_W && gy >= 0 && gy < IMG_H) v = eb[(size_t)gy * IMG_W + gx];
    s_e[t] = v;
  }
  for (int t = threadIdx.x; t < 64 * 64; t += 256) {
    const int gx = tx0 + (t & 63);
    const int gy = ty0 + (t >> 6);
    const bool interior = (gx >= 1) && (gx < IMG_W - 1) && (gy >= 1) && (gy < IMG_H - 1);
    s_low[t] = (interior && lowm[(size_t)b * HW + (size_t)gy * IMG_W + gx]) ? 1 : 0;
  }
  if (threadIdx.x == 0) s_changed = 0;
  __syncthreads();
  for (int sweep = 0; sweep < 256; ++sweep) {
    int local = 0;
    for (int t = threadIdx.x; t < 64 * 64; t += 256) {
      const int c = ((t >> 6) + 1) * 66 + (t & 63) + 1;
      if (!s_e[c] && s_low[t]) {
        const int any = s_e[c - 67] | s_e[c - 66] | s_e[c - 65] |
                        s_e[c - 1] | s_e[c + 1] |
                        s_e[c + 65] | s_e[c + 66] | s_e[c + 67];
        if (any) { s_e[c] = 1; local = 1; }
      }
    }
    if (local) atomicOr(&s_changed, 1);
    __syncthreads();
    const int ch = s_changed;
    __syncthreads();
    if (threadIdx.x == 0) s_changed = 0;
    if (!ch) break;
    __syncthreads();
  }
  __syncthreads();
  for (int t = threadIdx.x; t < 64 * 64; t += 256) {
    const int gx = tx0 + (t & 63);
    const int gy = ty0 + (t >> 6);
    eout[(size_t)b * HW + (size_t)gy * IMG_W + gx] = s_e[((t >> 6) + 1) * 66 + (t & 63) + 1];
  }
}

// ---------------------------------------------------------------------------
// Pass 6: out = 1 - edge
// ---------------------------------------------------------------------------
__global__ __launch_bounds__(256) void k_final(const unsigned char* __restrict__ e,
                                               float* __restrict__ out) {
  const size_t i = (size_t)blockIdx.x * 256 + threadIdx.x;
  out[i] = 1.0f - (float)e[i];
}

extern "C" void kernel_launch(void* const* d_in, const int* in_sizes, int n_in,
                              void* d_out, int out_size, void* d_ws, size_t ws_size,
                              hipStream_t stream) {
  (void)in_sizes; (void)n_in; (void)out_size; (void)ws_size;
  const float* x = (const float*)d_in[0];
  const float* gk = (const float*)d_in[1];  // 5x5 gaussian (sobel taps hardcoded)
  float* out = (float*)d_out;

  const size_t N = (size_t)IMG_B * HW;  // 8 Mi pixels
  float* fA = (float*)d_ws;             // 32 MB
  float* fB = fA + N;                   // 32 MB
  // after blur2 lands in fA, fB's 32 MB is reused for the three 8 MB byte masks
  unsigned char* e0 = (unsigned char*)fB;
  unsigned char* e1 = e0 + N;
  unsigned char* low = e1 + N;

  k_gray<<<(unsigned)(N / 256), 256, 0, stream>>>(x, fA);

  dim3 cg(IMG_W / 64, IMG_H / 16, IMG_B);
  k_conv5<<<cg, 256, 0, stream>>>(fA, gk, fB, 0);  // conv with K
  k_conv5<<<cg, 256, 0, stream>>>(fB, gk, fA, 1);  // conv with K^T
  k_sobel_nms<<<cg, 256, 0, stream>>>(fA, e0, low);

  dim3 hg(IMG_W / 64, IMG_H / 64, IMG_B);
  for (int i = 0; i < 16; ++i) {  // fixed count: deterministic under graph replay
    const unsigned char* src = (i & 1) ? e1 : e0;
    unsigned char* dst = (i & 1) ? e0 : e1;
    k_hyst<<<hg, 256, 0, stream>>>(src, low, dst);
  }
  k_final<<<(unsigned)(N / 256), 256, 0, stream>>>(e0, out);
}